// selfatt_46926812676531
// MI455X (gfx1250) — compile-verified
//
#include <hip/hip_runtime.h>

typedef __attribute__((ext_vector_type(2))) float v2f;
typedef __attribute__((ext_vector_type(8))) float v8f;

#define B_    8
#define T_    2048
#define C_    1024
#define M_    128
#define SCALE 0.03125f        // 1/sqrt(1024)

// LDS layout (floats). Padded strides to avoid bank conflicts on the
// per-lane float2 fragment reads.
#define QS_STRIDE  1028
#define ATT_STRIDE 274
#define QS_OFF     0
#define ATT_OFF    (16 * QS_STRIDE)                 // 16448
#define PSUM_OFF   (ATT_OFF + 16 * ATT_STRIDE)      // +4384
#define INV_OFF    (PSUM_OFF + 256)
#define WMAX_OFF   (INV_OFF + 16)
#define FROW_OFF   (WMAX_OFF + 128)
#define LDS_FLOATS (FROW_OFF + 16)                  // ~21.3K floats = ~85 KB

__global__ __launch_bounds__(256)
void band_attn_kernel(const float* __restrict__ Q,
                      const float* __restrict__ K,
                      const float* __restrict__ V,
                      const float* __restrict__ Msk,
                      float* __restrict__ ws)
{
    extern __shared__ float lds[];

    const int tile = blockIdx.x;         // 0..127 (t-tile)
    const int b    = blockIdx.y;         // 0..7
    const int t0   = tile * 16;
    const int tid  = threadIdx.x;
    const int lane = tid & 31;
    const int wave = tid >> 5;           // 0..7
    const int ln16 = lane & 15;
    const int hh   = lane >> 4;          // 0 or 1 (half-wave)

    const size_t bofs = (size_t)b * T_ * C_;

    // ---- Stage Q tile (contiguous 16*1024 fp32 = 64 KB) into LDS ----
    {
        const float4* src = (const float4*)(Q + bofs + (size_t)t0 * C_);
        for (int i = tid; i < (16 * C_) / 4; i += 256) {
            float4 q4 = src[i];
            int f = i * 4;
            int r = f >> 10;
            int c = f & (C_ - 1);
            float* dst = &lds[QS_OFF + r * QS_STRIDE + c];
            dst[0] = q4.x; dst[1] = q4.y; dst[2] = q4.z; dst[3] = q4.w;
        }
    }
    __syncthreads();

    // Band extent for this t-tile (16-aligned by construction: m=128, T=2048)
    const int sBase = (t0 - M_) < 0 ? 0 : (t0 - M_);
    int sEnd = t0 + 16 + M_; if (sEnd > T_) sEnd = T_;
    const int nS     = sEnd - sBase;     // <= 272, multiple of 16
    const int nTiles = nS >> 4;          // <= 17

    // ---- Phase A: energy tiles (Q @ K^T), mask, exp -> LDS att ----
    // fp32 WMMA 16x16x4: A/B fragment = float2 per lane:
    //   lane L, vgpr j -> element [row ln16][k0 + 2*hh + j]  (same for A and B^T)
    const int qbase = ln16 * QS_STRIDE + hh * 2;
    for (int j = wave; j < nTiles; j += 8) {
        const int s0 = sBase + j * 16;
        const float* kb = K + bofs + (size_t)(s0 + ln16) * C_ + hh * 2;
        v8f acc = {0.f, 0.f, 0.f, 0.f, 0.f, 0.f, 0.f, 0.f};
        #pragma unroll 8
        for (int k0 = 0; k0 < C_; k0 += 4) {
            v2f a  = *(const v2f*)(&lds[QS_OFF + qbase + k0]);
            v2f bb = *(const v2f*)(kb + k0);
            acc = __builtin_amdgcn_wmma_f32_16x16x4_f32(
                false, a, false, bb, (short)0, acc, false, false);
        }
        // C/D layout: lane L, vgpr v -> (M = v + 8*hh, N = ln16)
        const int s    = s0 + ln16;
        const int srel = (s0 - sBase) + ln16;
        #pragma unroll
        for (int v = 0; v < 8; ++v) {
            int t = t0 + v + 8 * hh;
            int d = t - s;
            float e = (d <= M_ && d >= -M_) ? __expf(acc[v] * SCALE) : 0.0f;
            lds[ATT_OFF + (v + 8 * hh) * ATT_STRIDE + srel] = e;
        }
    }
    __syncthreads();

    // ---- Row sums of exp -> 1/rowsum ----
    {
        int row = tid & 15;
        int chunk = tid >> 4;            // 16 chunks of 17 columns cover 272
        float p = 0.f;
        #pragma unroll
        for (int q = 0; q < 17; ++q) {
            int col = chunk * 17 + q;
            if (col < nS) p += lds[ATT_OFF + row * ATT_STRIDE + col];
        }
        lds[PSUM_OFF + row * 16 + chunk] = p;
    }
    __syncthreads();
    if (tid < 16) {
        float ssum = 0.f;
        #pragma unroll
        for (int q = 0; q < 16; ++q) ssum += lds[PSUM_OFF + tid * 16 + q];
        lds[INV_OFF + tid] = 1.0f / ssum;
    }
    __syncthreads();

    // ---- Phase B: lc = att @ V_band, running max over channels ----
    const int abase = ln16 * ATT_STRIDE + hh * 2;
    v8f macc = {-3.0e38f, -3.0e38f, -3.0e38f, -3.0e38f,
                -3.0e38f, -3.0e38f, -3.0e38f, -3.0e38f};
    for (int i = 0; i < 8; ++i) {
        const int ct = wave + i * 8;     // c-tile 0..63, uniform per wave
        const int c0 = ct * 16;
        // B fragment: lane L, vgpr j -> V[sBase + k0 + 2*hh + j][c0 + ln16]
        const float* vb = V + bofs + (size_t)(sBase + hh * 2) * C_ + c0 + ln16;
        v8f acc = {0.f, 0.f, 0.f, 0.f, 0.f, 0.f, 0.f, 0.f};
        #pragma unroll 4
        for (int k0 = 0; k0 < nS; k0 += 4) {
            v2f a = *(const v2f*)(&lds[ATT_OFF + abase + k0]);
            v2f bb;
            bb.x = vb[(size_t)(k0 + 0) * C_];
            bb.y = vb[(size_t)(k0 + 1) * C_];
            if (k0 + 8 < nS)
                __builtin_prefetch(vb + (size_t)(k0 + 8) * C_, 0, 0);
            acc = __builtin_amdgcn_wmma_f32_16x16x4_f32(
                false, a, false, bb, (short)0, acc, false, false);
        }
        #pragma unroll
        for (int v = 0; v < 8; ++v)
            macc[v] = fmaxf(macc[v], acc[v]);
    }

    // ---- Max over N within each 16-lane half (xor masks 1..8 keep halves) ----
    #pragma unroll
    for (int v = 0; v < 8; ++v) {
        float x = macc[v];
        #pragma unroll
        for (int off = 1; off < 16; off <<= 1)
            x = fmaxf(x, __shfl_xor(x, off, 32));
        if (ln16 == 0)
            lds[WMAX_OFF + (v + 8 * hh) * 8 + wave] = x;  // [row][wave]
    }
    __syncthreads();

    // ---- Combine: mask * invsum * global row max; sum rows (fixed order) ----
    if (tid < 16) {
        float g = lds[WMAX_OFF + tid * 8 + 0];
        #pragma unroll
        for (int w = 1; w < 8; ++w)
            g = fmaxf(g, lds[WMAX_OFF + tid * 8 + w]);
        float mv = Msk[(size_t)b * T_ + t0 + tid];
        lds[FROW_OFF + tid] = mv * lds[INV_OFF + tid] * g;
    }
    __syncthreads();
    if (tid == 0) {
        float s = 0.f;
        #pragma unroll
        for (int q = 0; q < 16; ++q) s += lds[FROW_OFF + q];
        ws[b * (T_ / 16) + tile] = s;
    }
}

// Deterministic tree reduction of 128 per-tile partials per batch -> out[b]
__global__ __launch_bounds__(128)
void band_attn_reduce(const float* __restrict__ ws, float* __restrict__ out)
{
    __shared__ float s[128];
    const int b = blockIdx.x;
    const int t = threadIdx.x;
    s[t] = ws[b * 128 + t];
    __syncthreads();
    for (int off = 64; off > 0; off >>= 1) {
        if (t < off) s[t] += s[t + off];
        __syncthreads();
    }
    if (t == 0) out[b] = s[0];
}

extern "C" void kernel_launch(void* const* d_in, const int* in_sizes, int n_in,
                              void* d_out, int out_size, void* d_ws, size_t ws_size,
                              hipStream_t stream)
{
    (void)in_sizes; (void)n_in; (void)out_size; (void)ws_size;
    const float* Q   = (const float*)d_in[0];
    const float* K   = (const float*)d_in[1];
    const float* V   = (const float*)d_in[2];
    const float* Msk = (const float*)d_in[3];
    // d_in[4] is m (=128), baked into the band geometry at compile time.
    float* out = (float*)d_out;
    float* ws  = (float*)d_ws;      // needs B_*128 floats = 4 KB

    dim3 grid(T_ / 16, B_);
    size_t shmem = (size_t)LDS_FLOATS * sizeof(float);
    band_attn_kernel<<<grid, 256, shmem, stream>>>(Q, K, V, Msk, ws);
    band_attn_reduce<<<B_, 128, 0, stream>>>(ws, out);
}